// VAENode_36996848288046
// MI455X (gfx1250) — compile-verified
//
#include <hip/hip_runtime.h>
#include <math.h>

// ---------------- problem constants (match reference) ----------------
#define BGR 256                 // graphs
#define NN  128                 // nodes per graph
#define PP  8128                // N*(N-1)/2 pairs per graph
#define EE  (256*1024)          // edges
#define DD  128                 // feature dim
#define LL  32                  // latent dim
#define BN  (BGR*NN)            // total nodes = 32768

typedef __attribute__((ext_vector_type(2))) float v2f;
typedef __attribute__((ext_vector_type(8))) float v8f;

__device__ __forceinline__ v8f wmma_f32_4(v2f a, v2f b, v8f c) {
  // D = A(16x4,f32) * B(4x16,f32) + C(16x16,f32)  ->  v_wmma_f32_16x16x4_f32
  return __builtin_amdgcn_wmma_f32_16x16x4_f32(false, a, false, b, (short)0, c,
                                               false, false);
}

__device__ __forceinline__ float wave_reduce_add(float v) {
  #pragma unroll
  for (int off = 16; off > 0; off >>= 1) v += __shfl_xor(v, off, 32);
  return v;
}

// ---------------- workspace layout (bytes) ----------------
// agg  : BN*DD f32   (16,777,216 B)
// h    : BN*DD f32   (16,777,216 B)
// z    : BN*LL f32   ( 4,194,304 B)
// kls  : BGR  f32
// logp : BGR  f32
// tgt  : BGR*PP u8   ( 2,080,768 B)
#define OFF_AGG  ((size_t)0)
#define OFF_H    (OFF_AGG + (size_t)BN*DD*4)
#define OFF_Z    (OFF_H   + (size_t)BN*DD*4)
#define OFF_KLS  (OFF_Z   + (size_t)BN*LL*4)
#define OFF_LOGP (OFF_KLS + (size_t)BGR*4)
#define OFF_TGT  (OFF_LOGP + (size_t)BGR*4)

// ---------------- kernel 0: zero scratch ----------------
__global__ void k_zero(float* p, long long nwords) {
  long long i = (long long)blockIdx.x * blockDim.x + threadIdx.x;
  long long stride = (long long)gridDim.x * blockDim.x;
  for (; i < nwords; i += stride) p[i] = 0.0f;
}

// ---------------- kernel 1: symmetric edge scatter + targets ----------------
__global__ void k_edge_scatter(const long long* __restrict__ ei,
                               const float* __restrict__ x,
                               float* __restrict__ agg,
                               unsigned char* __restrict__ tgt) {
  int e = blockIdx.x * (blockDim.x >> 5) + (threadIdx.x >> 5); // 1 wave / edge
  int lane = threadIdx.x & 31;
  long long src = ei[e];
  long long dst = ei[(size_t)EE + e];

  const float4 vs = ((const float4*)(x + (size_t)src * DD))[lane];
  const float4 vd = ((const float4*)(x + (size_t)dst * DD))[lane];
  float* as = agg + (size_t)src * DD + lane * 4;
  float* ad = agg + (size_t)dst * DD + lane * 4;
  atomicAdd(ad + 0, vs.x); atomicAdd(ad + 1, vs.y);
  atomicAdd(ad + 2, vs.z); atomicAdd(ad + 3, vs.w);
  atomicAdd(as + 0, vd.x); atomicAdd(as + 1, vd.y);
  atomicAdd(as + 2, vd.z); atomicAdd(as + 3, vd.w);

  if (lane == 0) {
    int li = (int)(src % NN), lj = (int)(dst % NN), b = (int)(src / NN);
    int pidx = li * (2 * NN - li - 1) / 2 + (lj - li - 1);
    tgt[(size_t)b * PP + pidx] = 1;   // .set() semantics; races write same value
  }
}

// ---------------- kernel 2: h = relu((x+agg) @ W1 + b1) ----------------
// grid: BN/16 blocks; 128 threads = 4 waves; wave w covers n-tiles {w, w+4}
__global__ void k_gemm1(const float* __restrict__ x,
                        const float* __restrict__ agg,
                        const float* __restrict__ W1,
                        const float* __restrict__ b1,
                        float* __restrict__ h) {
  const int tileM = blockIdx.x;
  const int wave  = threadIdx.x >> 5;
  const int lane  = threadIdx.x & 31;
  const int l15   = lane & 15;
  const int kb    = (lane >> 4) * 2;        // 0 for lanes 0-15, 2 for 16-31

  const int mrow = tileM * 16 + l15;        // A fragment row (M = lane&15)
  const int n0   = wave * 16 + l15;
  const int n1   = (wave + 4) * 16 + l15;

  v8f c0 = {}; v8f c1 = {};
  const float* xr = x   + (size_t)mrow * DD;
  const float* ar = agg + (size_t)mrow * DD;

  #pragma unroll 4
  for (int k0 = 0; k0 < DD; k0 += 4) {
    const int ka = k0 + kb;
    v2f a;
    a.x = xr[ka]     + ar[ka];
    a.y = xr[ka + 1] + ar[ka + 1];
    v2f bf0, bf1;
    bf0.x = W1[(size_t)ka * DD + n0];  bf0.y = W1[(size_t)(ka + 1) * DD + n0];
    bf1.x = W1[(size_t)ka * DD + n1];  bf1.y = W1[(size_t)(ka + 1) * DD + n1];
    c0 = wmma_f32_4(a, bf0, c0);
    c1 = wmma_f32_4(a, bf1, c1);
  }

  const float bias0 = b1[n0], bias1 = b1[n1];
  #pragma unroll
  for (int v = 0; v < 8; ++v) {
    const int row = tileM * 16 + ((lane >= 16) ? (v + 8) : v);
    h[(size_t)row * DD + n0] = fmaxf(c0[v] + bias0, 0.0f);
    h[(size_t)row * DD + n1] = fmaxf(c1[v] + bias1, 0.0f);
  }
}

// ---------------- kernel 3: mu/log_std/z + per-graph KL ----------------
// grid: BN/16/4 blocks; 128 threads = 4 waves; wave handles one 16-row tile,
// both 16-col tiles of L=32, for both mu and log_std (shared A fragment).
__global__ void k_latent(const float* __restrict__ h,
                         const float* __restrict__ Wmu,
                         const float* __restrict__ bmu,
                         const float* __restrict__ Wls,
                         const float* __restrict__ bls,
                         const float* __restrict__ eps,
                         float* __restrict__ z,
                         float* __restrict__ kls) {
  const int wave = threadIdx.x >> 5;
  const int lane = threadIdx.x & 31;
  const int l15  = lane & 15;
  const int kb   = (lane >> 4) * 2;
  const int tileM = blockIdx.x * 4 + wave;

  const int mrow = tileM * 16 + l15;
  v8f cmu0 = {}; v8f cmu1 = {}; v8f cls0 = {}; v8f cls1 = {};
  const float* hr = h + (size_t)mrow * DD;
  const int n0 = l15, n1 = 16 + l15;

  #pragma unroll 4
  for (int k0 = 0; k0 < DD; k0 += 4) {
    const int ka = k0 + kb;
    v2f a; a.x = hr[ka]; a.y = hr[ka + 1];
    v2f bm0, bm1, bl0, bl1;
    bm0.x = Wmu[(size_t)ka * LL + n0]; bm0.y = Wmu[(size_t)(ka + 1) * LL + n0];
    bm1.x = Wmu[(size_t)ka * LL + n1]; bm1.y = Wmu[(size_t)(ka + 1) * LL + n1];
    bl0.x = Wls[(size_t)ka * LL + n0]; bl0.y = Wls[(size_t)(ka + 1) * LL + n0];
    bl1.x = Wls[(size_t)ka * LL + n1]; bl1.y = Wls[(size_t)(ka + 1) * LL + n1];
    cmu0 = wmma_f32_4(a, bm0, cmu0);
    cmu1 = wmma_f32_4(a, bm1, cmu1);
    cls0 = wmma_f32_4(a, bl0, cls0);
    cls1 = wmma_f32_4(a, bl1, cls1);
  }

  float acc = 0.0f;
  const float bm_0 = bmu[n0], bm_1 = bmu[n1], bl_0 = bls[n0], bl_1 = bls[n1];
  #pragma unroll
  for (int v = 0; v < 8; ++v) {
    const int row = tileM * 16 + ((lane >= 16) ? (v + 8) : v);
    // n-tile 0
    {
      const float mu = cmu0[v] + bm_0;
      const float ls = cls0[v] + bl_0;
      const float sd = expf(ls);
      z[(size_t)row * LL + n0] = mu + sd * eps[(size_t)row * LL + n0];
      acc += sd * sd + mu * mu - 1.0f - 2.0f * ls;
    }
    // n-tile 1
    {
      const float mu = cmu1[v] + bm_1;
      const float ls = cls1[v] + bl_1;
      const float sd = expf(ls);
      z[(size_t)row * LL + n1] = mu + sd * eps[(size_t)row * LL + n1];
      acc += sd * sd + mu * mu - 1.0f - 2.0f * ls;
    }
  }
  acc = wave_reduce_add(acc);
  if (lane == 0) atomicAdd(&kls[tileM / 8], 0.5f * acc);  // 8 tiles per graph
}

// ---------------- kernel 4: decoder  logits = Z Z^T, masked ELBO term -------
// grid: BGR blocks; 256 threads = 8 waves; 36 upper-tri 16x16 tiles per graph.
__global__ void k_decoder(const float* __restrict__ z,
                          const unsigned char* __restrict__ tgt,
                          float* __restrict__ logp) {
  const int b    = blockIdx.x;
  const int wave = threadIdx.x >> 5;
  const int lane = threadIdx.x & 31;
  const int l15  = lane & 15;
  const int kb   = (lane >> 4) * 2;

  const float* zb = z + (size_t)b * NN * LL;
  const unsigned char* tb = tgt + (size_t)b * PP;

  float acc = 0.0f;
  for (int t = wave; t < 36; t += 8) {       // uniform per wave -> EXEC all-1s
    // map t -> (ti, tj), ti <= tj, 8x8 tile grid upper triangle
    int ti = 0, tj = 0, cnt = 0;
    #pragma unroll
    for (int aa = 0; aa < 8; ++aa)
      #pragma unroll
      for (int bb = aa; bb < 8; ++bb) {
        if (cnt == t) { ti = aa; tj = bb; }
        ++cnt;
      }

    const float* ra = zb + (size_t)(ti * 16 + l15) * LL;  // A rows (i)
    const float* rb = zb + (size_t)(tj * 16 + l15) * LL;  // B cols (j) = A-style
    v8f c = {};
    #pragma unroll
    for (int k0 = 0; k0 < LL; k0 += 4) {
      const int ka = k0 + kb;
      v2f a;  a.x = ra[ka]; a.y = ra[ka + 1];
      v2f bf; bf.x = rb[ka]; bf.y = rb[ka + 1];
      c = wmma_f32_4(a, bf, c);
    }

    #pragma unroll
    for (int v = 0; v < 8; ++v) {
      const int i = ti * 16 + ((lane >= 16) ? (v + 8) : v);
      const int j = tj * 16 + l15;
      if (i < j) {
        const float g = c[v];
        const int pidx = i * (2 * NN - i - 1) / 2 + (j - i - 1);
        const float sp = (g > 0.0f) ? (g + log1pf(expf(-g))) : log1pf(expf(g));
        acc += (tb[pidx] ? g : 0.0f) - sp;
      }
    }
  }
  acc = wave_reduce_add(acc);
  if (lane == 0) atomicAdd(&logp[b], acc);
}

// ---------------- kernel 5: -mean(logp - kls) ----------------
__global__ void k_finalize(const float* __restrict__ kls,
                           const float* __restrict__ logp,
                           float* __restrict__ out) {
  __shared__ float sm[BGR];
  const int t = threadIdx.x;
  sm[t] = logp[t] - kls[t];
  __syncthreads();
  for (int s = BGR / 2; s > 0; s >>= 1) {
    if (t < s) sm[t] += sm[t + s];
    __syncthreads();
  }
  if (t == 0) out[0] = -sm[0] / (float)BGR;
}

// ---------------- host launch ----------------
extern "C" void kernel_launch(void* const* d_in, const int* in_sizes, int n_in,
                              void* d_out, int out_size, void* d_ws, size_t ws_size,
                              hipStream_t stream) {
  const float*     x    = (const float*)d_in[0];
  const long long* ei   = (const long long*)d_in[1];
  (void)d_in[2];                                   // batch (implied by layout)
  const float*     eps  = (const float*)d_in[3];
  const float*     W1   = (const float*)d_in[4];
  const float*     b1   = (const float*)d_in[5];
  const float*     Wmu  = (const float*)d_in[6];
  const float*     bmu  = (const float*)d_in[7];
  const float*     Wls  = (const float*)d_in[8];
  const float*     bls  = (const float*)d_in[9];
  (void)in_sizes; (void)n_in; (void)out_size; (void)ws_size;

  char* ws = (char*)d_ws;
  float*         agg  = (float*)(ws + OFF_AGG);
  float*         h    = (float*)(ws + OFF_H);
  float*         z    = (float*)(ws + OFF_Z);
  float*         kls  = (float*)(ws + OFF_KLS);
  float*         logp = (float*)(ws + OFF_LOGP);
  unsigned char* tgt  = (unsigned char*)(ws + OFF_TGT);

  // zero agg, and the contiguous [kls | logp | tgt] region
  const long long nAgg = (long long)BN * DD;
  const long long nTail = (long long)(2 * BGR) + ((long long)BGR * PP) / 4;
  k_zero<<<2048, 256, 0, stream>>>(agg, nAgg);
  k_zero<<<512, 256, 0, stream>>>(kls, nTail);

  k_edge_scatter<<<EE / 8, 256, 0, stream>>>(ei, x, agg, tgt);
  k_gemm1<<<BN / 16, 128, 0, stream>>>(x, agg, W1, b1, h);
  k_latent<<<BN / 64, 128, 0, stream>>>(h, Wmu, bmu, Wls, bls, eps, z, kls);
  k_decoder<<<BGR, 256, 0, stream>>>(z, tgt, logp);
  k_finalize<<<1, BGR, 0, stream>>>(kls, logp, (float*)d_out);
}